// MambaTimeSeriesModel_16887811408524
// MI455X (gfx1250) — compile-verified
//
#include <hip/hip_runtime.h>
#include <math.h>

#define D_MODEL 512
#define N_LAYERS 4
#define D_INNER 1024
#define D_STATE 16
#define D_CONV 4
#define DT_RANK 32
#define B_SZ 16
#define L_IN 512
#define L_OUT 24

typedef float v2f __attribute__((ext_vector_type(2)));
typedef float v8f __attribute__((ext_vector_type(8)));

__device__ __forceinline__ float siluf(float v) { return v / (1.0f + __expf(-v)); }
__device__ __forceinline__ float softplusf(float v) { return (v > 20.0f) ? v : log1pf(__expf(v)); }

// Load one K=8 chunk (2 k-steps of 4) for A and the 4 B sub-tiles.
__device__ __forceinline__ void load_chunk(const float* __restrict__ Ar,
                                           const float* __restrict__ W0,
                                           const float* __restrict__ W1,
                                           const float* __restrict__ W2,
                                           const float* __restrict__ W3,
                                           int kk, v2f (&a)[2], v2f (&b)[4][2]) {
#pragma unroll
  for (int t = 0; t < 2; ++t) {
    const int o = kk + 4 * t;
    a[t]    = *(const v2f*)(Ar + o);
    b[0][t] = *(const v2f*)(W0 + o);
    b[1][t] = *(const v2f*)(W1 + o);
    b[2][t] = *(const v2f*)(W2 + o);
    b[3][t] = *(const v2f*)(W3 + o);
  }
}

__device__ __forceinline__ void mma_chunk(const v2f (&a)[2], const v2f (&b)[4][2],
                                          v8f& acc0, v8f& acc1, v8f& acc2, v8f& acc3) {
#pragma unroll
  for (int t = 0; t < 2; ++t) {
    acc0 = __builtin_amdgcn_wmma_f32_16x16x4_f32(false, a[t], false, b[0][t], (short)0, acc0, false, false);
    acc1 = __builtin_amdgcn_wmma_f32_16x16x4_f32(false, a[t], false, b[1][t], (short)0, acc1, false, false);
    acc2 = __builtin_amdgcn_wmma_f32_16x16x4_f32(false, a[t], false, b[2][t], (short)0, acc2, false, false);
    acc3 = __builtin_amdgcn_wmma_f32_16x16x4_f32(false, a[t], false, b[3][t], (short)0, acc3, false, false);
  }
}

// ---------------------------------------------------------------------------
// Generic NT GEMM on the CDNA5 matrix pipe:
//   C[M,N] = act( A[M,K] @ W[N,K]^T + bias + Cinit )
// One wave (32 lanes) computes a 16(M) x 64(N) tile with v_wmma_f32_16x16x4_f32.
// K staged in 8-wide chunks, double-buffered, tail peeled so the steady-state
// loop body is branch-free (no spills, loads issued one chunk ahead).
// A lane layout (16x4 f32): lane%16 = M row, VGPR pair = K{2*(lane/16), +1}.
// B lane layout (4x16 f32): lane%16 = N col, VGPR pair = K{2*(lane/16), +1}.
// C/D layout: VGPR v -> (M = v + 8*(lane/16), N = lane%16).
// Requires M%16==0, N%64==0, K%16==0 (true for all call sites here).
// ---------------------------------------------------------------------------
__global__ void k_gemm_nt(const float* __restrict__ A, int lda,
                          const float* __restrict__ W,      // (N,K) row-major
                          const float* __restrict__ Cinit,  // optional
                          const float* __restrict__ bias,   // optional, per-col
                          float* __restrict__ C, int ldc,
                          int M, int N, int K, int act) {
  const int lane  = threadIdx.x;
  const int l16   = lane & 15;
  const int half  = lane >> 4;
  const int ntile = blockIdx.x * 64;
  const int mtile = blockIdx.y * 16;

  const float* Ar = A + (size_t)(mtile + l16) * lda + 2 * half;
  const float* W0 = W + (size_t)(ntile +  0 + l16) * K + 2 * half;
  const float* W1 = W + (size_t)(ntile + 16 + l16) * K + 2 * half;
  const float* W2 = W + (size_t)(ntile + 32 + l16) * K + 2 * half;
  const float* W3 = W + (size_t)(ntile + 48 + l16) * K + 2 * half;

  v8f acc0 = {}, acc1 = {}, acc2 = {}, acc3 = {};
  v2f aA[2], bA[4][2];   // buffer set A
  v2f aB[2], bB[4][2];   // buffer set B

  load_chunk(Ar, W0, W1, W2, W3, 0, aA, bA);
  int k = 0;
  for (; k < K - 16; k += 16) {
    load_chunk(Ar, W0, W1, W2, W3, k + 8, aB, bB);
    mma_chunk(aA, bA, acc0, acc1, acc2, acc3);
    load_chunk(Ar, W0, W1, W2, W3, k + 16, aA, bA);
    mma_chunk(aB, bB, acc0, acc1, acc2, acc3);
  }
  // peeled tail: chunks k and k+8
  load_chunk(Ar, W0, W1, W2, W3, k + 8, aB, bB);
  mma_chunk(aA, bA, acc0, acc1, acc2, acc3);
  mma_chunk(aB, bB, acc0, acc1, acc2, acc3);

#pragma unroll
  for (int v = 0; v < 8; ++v) {
    const int row = mtile + v + 8 * half;
    float r[4] = { acc0[v], acc1[v], acc2[v], acc3[v] };
#pragma unroll
    for (int s = 0; s < 4; ++s) {
      const int col = ntile + s * 16 + l16;
      float val = r[s];
      if (Cinit) val += Cinit[(size_t)row * ldc + col];
      if (bias)  val += bias[col];
      if (act == 1) val = softplusf(val);
      C[(size_t)row * ldc + col] = val;
    }
  }
}

// h[r,d] = x[r] * w[d] + b[d]     (input projection, K = 1)
__global__ void k_inproj(const float* __restrict__ x, const float* __restrict__ w,
                         const float* __restrict__ b, float* __restrict__ out, int rows) {
  int idx = blockIdx.x * blockDim.x + threadIdx.x;
  int total = rows * D_MODEL;
  if (idx >= total) return;
  int r = idx / D_MODEL, d = idx - r * D_MODEL;
  out[idx] = x[r] * w[d] + b[d];
}

// Wave-per-row RMSNorm over D_MODEL=512 (16 elements/lane, wave32 butterfly).
__global__ void k_rmsnorm(const float* __restrict__ x, const float* __restrict__ w,
                          float* __restrict__ out, int rows) {
  int row = blockIdx.x;
  int lane = threadIdx.x;
  const float* xr = x + (size_t)row * D_MODEL;
  float ss = 0.0f;
#pragma unroll
  for (int i = 0; i < D_MODEL / 32; ++i) { float v = xr[lane + 32 * i]; ss += v * v; }
#pragma unroll
  for (int off = 16; off > 0; off >>= 1) ss += __shfl_xor(ss, off, 32);
  float scale = rsqrtf(ss * (1.0f / (float)D_MODEL) + 1e-5f);
  float* orow = out + (size_t)row * D_MODEL;
#pragma unroll
  for (int i = 0; i < D_MODEL / 32; ++i) {
    int c = lane + 32 * i;
    orow[c] = xr[c] * scale * w[c];
  }
}

// Causal depthwise conv (width 4, left pad 3) + bias + SiLU.
// xz is (rows, 2*D_INNER); channel part is the first D_INNER columns.
__global__ void k_conv_silu(const float* __restrict__ xz, const float* __restrict__ cw,
                            const float* __restrict__ cb, float* __restrict__ out,
                            int Bn, int L) {
  int idx = blockIdx.x * blockDim.x + threadIdx.x;
  int total = Bn * L * D_INNER;
  if (idx >= total) return;
  int d = idx % D_INNER;
  int l = (idx / D_INNER) % L;
  int b = idx / (D_INNER * L);
  float acc = cb[d];
#pragma unroll
  for (int k = 0; k < D_CONV; ++k) {
    int ls = l + k - (D_CONV - 1);
    if (ls >= 0) acc += xz[(size_t)(b * L + ls) * (2 * D_INNER) + d] * cw[d * D_CONV + k];
  }
  out[(size_t)(b * L + l) * D_INNER + d] = siluf(acc);
}

// Selective scan: one thread per (b,d) channel, 16 f32 states in registers.
// Blocks are aligned so all 256 threads share one batch b -> the dbl (B/C)
// row pointer is wave-uniform and lowers to scalar broadcast loads.
// delta buffer is read then overwritten with y (+ u*D) in place.
// grid.x must be Bn * (D_INNER/256), blockDim 256.
__global__ void k_scan(const float* __restrict__ u, float* __restrict__ delta,
                       const float* __restrict__ dbl, const float* __restrict__ alog,
                       const float* __restrict__ Dp, int L) {
  const int b = blockIdx.x >> 2;                       // 4 blocks per batch
  const int d = ((blockIdx.x & 3) << 8) + threadIdx.x; // 0..1023
  float A[D_STATE];
#pragma unroll
  for (int s = 0; s < D_STATE; ++s) A[s] = -expf(alog[d * D_STATE + s]);
  const float Dd = Dp[d];
  float h[D_STATE];
#pragma unroll
  for (int s = 0; s < D_STATE; ++s) h[s] = 0.0f;
  for (int l = 0; l < L; ++l) {
    const size_t r = (size_t)(b * L + l);
    float dt = delta[r * D_INNER + d];
    float ut = u[r * D_INNER + d];
    float du = dt * ut;
    const float* bc = dbl + r * 64;   // uniform across the block
    float y = ut * Dd;
#pragma unroll
    for (int s = 0; s < D_STATE; ++s) {
      h[s] = __expf(dt * A[s]) * h[s] + du * bc[DT_RANK + s];
      y += h[s] * bc[DT_RANK + D_STATE + s];
    }
    delta[r * D_INNER + d] = y;
  }
}

// gated[r,d] = y[r,d] * silu(res[r,d]); y lives in DELTA, res in xz[:,1024:].
__global__ void k_gate(float* __restrict__ y, const float* __restrict__ xz, int total) {
  int idx = blockIdx.x * blockDim.x + threadIdx.x;
  if (idx >= total) return;
  int r = idx / D_INNER, d = idx - r * D_INNER;
  float res = xz[(size_t)r * (2 * D_INNER) + D_INNER + d];
  y[idx] = y[idx] * siluf(res);
}

// Extract last timestep of prefill hidden as decode seed.
__global__ void k_copy_last(const float* __restrict__ xn, float* __restrict__ x) {
  int idx = blockIdx.x * blockDim.x + threadIdx.x;
  if (idx >= B_SZ * D_MODEL) return;
  int b = idx / D_MODEL, d = idx - b * D_MODEL;
  x[idx] = xn[(size_t)(b * L_IN + (L_IN - 1)) * D_MODEL + d];
}

// Output head: pred[b] = dot(xn[b,:], opw) + opb; write d_out and PRED.
__global__ void k_head(const float* __restrict__ xn, const float* __restrict__ opw,
                       const float* __restrict__ opb, float* __restrict__ dout,
                       float* __restrict__ pred, int step) {
  int b = blockIdx.x;
  int lane = threadIdx.x;
  const float* xr = xn + (size_t)b * D_MODEL;
  float s = 0.0f;
#pragma unroll
  for (int i = 0; i < D_MODEL / 32; ++i) s += xr[lane + 32 * i] * opw[lane + 32 * i];
#pragma unroll
  for (int off = 16; off > 0; off >>= 1) s += __shfl_xor(s, off, 32);
  if (lane == 0) {
    float p = s + opb[0];
    dout[b * L_OUT + step] = p;
    pred[b] = p;
  }
}

// ---------------------------------------------------------------------------

extern "C" void kernel_launch(void* const* d_in, const int* in_sizes, int n_in,
                              void* d_out, int out_size, void* d_ws, size_t ws_size,
                              hipStream_t stream) {
  (void)in_sizes; (void)n_in; (void)out_size; (void)ws_size;
  const float* x_in  = (const float*)d_in[0];
  const float* ipw   = (const float*)d_in[1];
  const float* ipb   = (const float*)d_in[2];
  const float* opw   = (const float*)d_in[3];
  const float* opb   = (const float*)d_in[4];
  const float* normw = (const float*)d_in[5];
  const float* inpw  = (const float*)d_in[6];
  const float* convw = (const float*)d_in[7];
  const float* convb = (const float*)d_in[8];
  const float* xpw   = (const float*)d_in[9];
  const float* dtw   = (const float*)d_in[10];
  const float* dtb   = (const float*)d_in[11];
  const float* alog  = (const float*)d_in[12];
  const float* Dp    = (const float*)d_in[13];
  const float* outw  = (const float*)d_in[14];
  const float* normf = (const float*)d_in[15];

  float* ws = (float*)d_ws;
  const size_t rmax = (size_t)B_SZ * L_IN;
  float* X     = ws;                           // residual stream  (rmax*512)
  float* XN    = X     + rmax * D_MODEL;       // normed input     (rmax*512)
  float* XZ    = XN    + rmax * D_MODEL;       // in_proj out      (rmax*2048)
  float* XCS   = XZ    + rmax * 2 * D_INNER;   // conv+silu (u)    (rmax*1024)
  float* DBL   = XCS   + rmax * D_INNER;       // dt|B|C           (rmax*64)
  float* DELTA = DBL   + rmax * 64;            // delta -> y -> gated (rmax*1024)
  float* PRED  = DELTA + rmax * D_INNER;       // decode preds     (16)

  auto mamba = [&](int L) {
    const int rows = B_SZ * L;
    for (int ly = 0; ly < N_LAYERS; ++ly) {
      k_rmsnorm<<<rows, 32, 0, stream>>>(X, normw + ly * D_MODEL, XN, rows);
      k_gemm_nt<<<dim3((2 * D_INNER) / 64, rows / 16), 32, 0, stream>>>(
          XN, D_MODEL, inpw + (size_t)ly * 2 * D_INNER * D_MODEL,
          nullptr, nullptr, XZ, 2 * D_INNER, rows, 2 * D_INNER, D_MODEL, 0);
      {
        int n = rows * D_INNER;
        k_conv_silu<<<(n + 255) / 256, 256, 0, stream>>>(
            XZ, convw + (size_t)ly * D_INNER * D_CONV, convb + (size_t)ly * D_INNER,
            XCS, B_SZ, L);
      }
      k_gemm_nt<<<dim3(64 / 64, rows / 16), 32, 0, stream>>>(
          XCS, D_INNER, xpw + (size_t)ly * 64 * D_INNER,
          nullptr, nullptr, DBL, 64, rows, 64, D_INNER, 0);
      k_gemm_nt<<<dim3(D_INNER / 64, rows / 16), 32, 0, stream>>>(
          DBL, 64, dtw + (size_t)ly * D_INNER * DT_RANK,
          nullptr, dtb + (size_t)ly * D_INNER, DELTA, D_INNER,
          rows, D_INNER, DT_RANK, 1);
      k_scan<<<B_SZ * (D_INNER / 256), 256, 0, stream>>>(
          XCS, DELTA, DBL, alog + (size_t)ly * D_INNER * D_STATE,
          Dp + (size_t)ly * D_INNER, L);
      {
        int n = rows * D_INNER;
        k_gate<<<(n + 255) / 256, 256, 0, stream>>>(DELTA, XZ, n);
      }
      k_gemm_nt<<<dim3(D_MODEL / 64, rows / 16), 32, 0, stream>>>(
          DELTA, D_INNER, outw + (size_t)ly * D_MODEL * D_INNER,
          X, nullptr, X, D_MODEL, rows, D_MODEL, D_INNER, 0);
    }
    k_rmsnorm<<<rows, 32, 0, stream>>>(X, normf, XN, rows);
  };

  // ---- prefill ----
  {
    int n = B_SZ * L_IN * D_MODEL;
    k_inproj<<<(n + 255) / 256, 256, 0, stream>>>(x_in, ipw, ipb, X, B_SZ * L_IN);
  }
  mamba(L_IN);
  k_copy_last<<<(B_SZ * D_MODEL + 255) / 256, 256, 0, stream>>>(XN, X);

  // ---- autoregressive decode (24 steps, L = 1) ----
  for (int step = 0; step < L_OUT; ++step) {
    mamba(1);
    k_head<<<B_SZ, 32, 0, stream>>>(XN, opw, opb, (float*)d_out, PRED, step);
    int n = B_SZ * D_MODEL;
    k_inproj<<<(n + 255) / 256, 256, 0, stream>>>(PRED, ipw, ipb, X, B_SZ);
  }
}